// XiLoss_55035710931109
// MI455X (gfx1250) — compile-verified
//
#include <hip/hip_runtime.h>
#include <stdint.h>

typedef __attribute__((ext_vector_type(2))) float v2f;
typedef __attribute__((ext_vector_type(8))) float v8f;

namespace {
constexpr int   THREADS = 256;          // 8 wave32 waves per block
constexpr int   TILE    = 4096;         // elements per sort tile
constexpr int   IPT     = TILE / THREADS;
constexpr float EPSN    = 1e-6f;
}

__device__ __forceinline__ uint32_t hmix(uint32_t x) {
  x ^= x >> 17; x *= 0xed5ad4bbu;
  x ^= x >> 11; x *= 0xac4c1b51u;
  x ^= x >> 15; x *= 0x31848babu;
  x ^= x >> 14;
  return x;
}
__device__ __forceinline__ float u01(uint32_t x) {
  return (float)(hmix(x) >> 8) * (1.0f / 16777216.0f);
}
// monotonic float -> uint key (ascending)
__device__ __forceinline__ uint32_t fkey(float f) {
  uint32_t u = __float_as_uint(f);
  return u ^ ((u & 0x80000000u) ? 0xFFFFFFFFu : 0x80000000u);
}
// LDS byte offset of a __shared__ object = low 32 bits of its flat address
__device__ __forceinline__ uint32_t lds_off32(const void* p) {
  return (uint32_t)(uintptr_t)p;
}

// 256-thread block exclusive scan (Hillis-Steele in LDS)
__device__ __forceinline__ unsigned bscan_excl(unsigned tsum, unsigned* lds, unsigned& total) {
  const int t = threadIdx.x;
  lds[t] = tsum;
  __syncthreads();
  for (int off = 1; off < THREADS; off <<= 1) {
    unsigned v = (t >= off) ? lds[t - off] : 0u;
    __syncthreads();
    lds[t] += v;
    __syncthreads();
  }
  unsigned incl = lds[t];
  total = lds[THREADS - 1];
  __syncthreads();
  return incl - tsum;
}

// ---------------- prep: noisy y_true, sortable keys, payload = index ----------------
__global__ void k_prep(const float* __restrict__ y_true, float* __restrict__ tn,
                       uint32_t* __restrict__ key, uint32_t* __restrict__ val, int n) {
  for (int i = blockIdx.x * blockDim.x + threadIdx.x; i < n; i += gridDim.x * blockDim.x) {
    float v = y_true[i] + (u01(0x51ed270bu + (uint32_t)i) - 0.5f) * EPSN;
    tn[i]  = v;
    key[i] = fkey(v);
    val[i] = (uint32_t)i;
  }
}

// ---------------- radix pass 1/3: per-tile digit histogram (b128 loads) ----------------
__global__ void k_hist(const uint32_t* __restrict__ key, uint32_t* __restrict__ hist,
                       int shift, int nblk) {
  __shared__ unsigned cnt[256];
  const int b = blockIdx.x, t = threadIdx.x;
  cnt[t] = 0u;
  __syncthreads();
  const int base = b * TILE;
  for (int s = 0; s < IPT / 4; ++s) {
    uint4 k4 = *(const uint4*)(key + base + (s * THREADS + t) * 4);
    atomicAdd(&cnt[(k4.x >> shift) & 0xFFu], 1u);
    atomicAdd(&cnt[(k4.y >> shift) & 0xFFu], 1u);
    atomicAdd(&cnt[(k4.z >> shift) & 0xFFu], 1u);
    atomicAdd(&cnt[(k4.w >> shift) & 0xFFu], 1u);
  }
  __syncthreads();
  hist[t * nblk + b] = cnt[t];   // digit-major for the scan
}

// ---------------- radix pass 2/3: global (digit,block) exclusive offsets ----------------
__global__ void k_scan(uint32_t* __restrict__ hist, int nblk) {
  __shared__ unsigned sb[THREADS];
  const int d = threadIdx.x;
  unsigned sum = 0;
  for (int b = 0; b < nblk; ++b) sum += hist[d * nblk + b];
  unsigned total;
  unsigned run = bscan_excl(sum, sb, total);   // digit base
  for (int b = 0; b < nblk; ++b) {
    unsigned c = hist[d * nblk + b];
    hist[d * nblk + b] = run;
    run += c;
  }
}

// ---------------- radix pass 3/3: async-staged stable tile sort + scatter ----------------
__global__ void k_scatter(const uint32_t* __restrict__ keyIn, const uint32_t* __restrict__ valIn,
                          uint32_t* __restrict__ keyOut, uint32_t* __restrict__ valOut,
                          const uint32_t* __restrict__ offs, int shift, int nblk) {
  __shared__ uint32_t keyS[TILE];      // async-DMA staged key tile (16 KB)
  __shared__ uint32_t tokA[TILE];
  __shared__ uint32_t tokB[TILE];
  __shared__ unsigned sb[THREADS];
  __shared__ unsigned localStart[256];
  const int b = blockIdx.x, t = threadIdx.x;
  const int base = b * TILE;

  // stage key tile into LDS via CDNA5 async DMA (tracked by ASYNCcnt)
  for (int r = 0; r < TILE / (THREADS * 4); ++r) {            // 4 rounds of b128/lane
    const int c4 = r * THREADS + t;                           // 16-byte chunk index
    uint32_t       loff  = lds_off32(&keyS[c4 * 4]);
    unsigned long long g = (unsigned long long)(uintptr_t)(keyIn + base + c4 * 4);
    asm volatile("global_load_async_to_lds_b128 %0, %1, off"
                 :: "v"(loff), "v"(g) : "memory");
  }
  asm volatile("s_wait_asynccnt 0x0" ::: "memory");
  __syncthreads();

  // token = digit[19:12] | tile position[11:0]
  for (int s = 0; s < IPT; ++s) {
    int idx = s * THREADS + t;
    uint32_t d = (keyS[idx] >> shift) & 0xFFu;
    tokA[idx] = (d << 12) | (uint32_t)idx;
  }
  __syncthreads();
  uint32_t* tin = tokA;
  uint32_t* tout = tokB;
  for (int bit = 0; bit < 8; ++bit) {
    uint32_t tok[IPT]; unsigned pre[IPT]; unsigned run = 0;
    for (int s = 0; s < IPT; ++s) {
      tok[s] = tin[t * IPT + s];
      unsigned f = ((tok[s] >> (12 + bit)) & 1u) ^ 1u;   // zeros first
      pre[s] = run; run += f;
    }
    unsigned totZ;
    unsigned excl = bscan_excl(run, sb, totZ);
    for (int s = 0; s < IPT; ++s) {
      unsigned pos = (unsigned)(t * IPT + s);
      unsigned f = ((tok[s] >> (12 + bit)) & 1u) ^ 1u;
      unsigned zr = excl + pre[s];
      unsigned np = f ? zr : (totZ + (pos - zr));
      tout[np] = tok[s];
    }
    __syncthreads();
    uint32_t* tmp = tin; tin = tout; tout = tmp;         // 8 swaps -> ends in tokA
  }
  // tile now stably sorted by digit; find digit group starts
  for (int s = 0; s < IPT; ++s) {
    int pos = t * IPT + s;
    uint32_t d  = tin[pos] >> 12;
    uint32_t dp = (pos == 0) ? 0x100u : (tin[pos - 1] >> 12);
    if (d != dp) localStart[d] = (unsigned)pos;
  }
  __syncthreads();
  for (int s = 0; s < IPT; ++s) {
    int pos = t * IPT + s;
    uint32_t tok  = tin[pos];
    uint32_t d    = tok >> 12;
    uint32_t orig = tok & 0xFFFu;
    uint32_t g = offs[d * (unsigned)nblk + (unsigned)b] + ((unsigned)pos - localStart[d]);
    keyOut[g] = keyS[orig];                 // LDS gather instead of global re-read
    valOut[g] = valIn[base + orig];
  }
}

// ---------------- gather y_pred through perm, re-key for second sort ----------------
__global__ void k_gather(const float* __restrict__ y_pred,
                         uint32_t* __restrict__ key, uint32_t* __restrict__ val, int n) {
  for (int p = blockIdx.x * blockDim.x + threadIdx.x; p < n; p += gridDim.x * blockDim.x) {
    uint32_t src = val[p];                       // perm from sort 1
    float v = y_pred[src] + (u01(0x9e3779b9u + (uint32_t)p) - 0.5f) * EPSN;
    key[p] = fkey(v);
    val[p] = (uint32_t)p;
  }
}

// ---------------- hard ranks: sorted position -> rank at original position ----------------
__global__ void k_rank(const uint32_t* __restrict__ val, uint32_t* __restrict__ rk, int n) {
  for (int q = blockIdx.x * blockDim.x + threadIdx.x; q < n; q += gridDim.x * blockDim.x)
    rk[val[q]] = (uint32_t)(q + 1);
}

// ---------------- WMMA f32 reduction engines: D = A(16x4) * ones(4x16) + C ----------------
// b128 loads, 2 independent accumulators -> 128 elements / wave / iteration
__global__ void k_mse(const float* __restrict__ a, const float* __restrict__ b,
                      float* __restrict__ part, int n) {
  const int lane = threadIdx.x & 31;
  const int wv   = threadIdx.x >> 5;
  const int wpb  = blockDim.x >> 5;
  const int gw   = blockIdx.x * wpb + wv;
  const int gwN  = gridDim.x * wpb;
  v8f acc0 = {}, acc1 = {};
  v2f onesv = {1.0f, 1.0f};
  for (long long bse = (long long)gw * 128; bse + 128 <= n; bse += (long long)gwN * 128) {
    long long i0 = bse + lane * 4;
    __builtin_prefetch(&a[i0 + (long long)gwN * 128], 0, 0);
    float4 va = *(const float4*)(a + i0);
    float4 vb = *(const float4*)(b + i0);
    float d0 = va.x - vb.x, d1 = va.y - vb.y, d2 = va.z - vb.z, d3 = va.w - vb.w;
    v2f x0 = {d0 * d0, d1 * d1};
    v2f x1 = {d2 * d2, d3 * d3};
    acc0 = __builtin_amdgcn_wmma_f32_16x16x4_f32(false, x0, false, onesv,
                                                 (short)0, acc0, false, false);
    acc1 = __builtin_amdgcn_wmma_f32_16x16x4_f32(false, x1, false, onesv,
                                                 (short)0, acc1, false, false);
  }
  v8f acc = acc0 + acc1;
  float s = acc[0] + acc[1] + acc[2] + acc[3] + acc[4] + acc[5] + acc[6] + acc[7];
  s += __shfl_xor(s, 16, 32);                    // rows 0-7 half + rows 8-15 half
  __shared__ float wsum[8];
  if (lane == 0) wsum[wv] = s;
  __syncthreads();
  if (threadIdx.x == 0) {
    float tsum = 0.f;
    for (int w = 0; w < wpb; ++w) tsum += wsum[w];
    part[blockIdx.x] = tsum;
  }
}

__global__ void k_diff(const uint32_t* __restrict__ rk, float* __restrict__ part, int n) {
  const int lane = threadIdx.x & 31;
  const int wv   = threadIdx.x >> 5;
  const int wpb  = blockDim.x >> 5;
  const int gw   = blockIdx.x * wpb + wv;
  const int gwN  = gridDim.x * wpb;
  v8f acc0 = {}, acc1 = {};
  v2f onesv = {1.0f, 1.0f};
  for (long long bse = (long long)gw * 128; bse + 128 <= n; bse += (long long)gwN * 128) {
    long long i0 = bse + lane * 4;
    __builtin_prefetch(&rk[i0 + (long long)gwN * 128], 0, 0);
    uint4 r = *(const uint4*)(rk + i0);
    long long inx = i0 + 4 < n ? i0 + 4 : (long long)n - 1;  // clamp => last diff is 0
    uint32_t r4 = rk[inx];
    float d0 = fabsf((float)((int)r.y - (int)r.x));
    float d1 = fabsf((float)((int)r.z - (int)r.y));
    float d2 = fabsf((float)((int)r.w - (int)r.z));
    float d3 = fabsf((float)((int)r4  - (int)r.w));
    v2f x0 = {d0, d1};
    v2f x1 = {d2, d3};
    acc0 = __builtin_amdgcn_wmma_f32_16x16x4_f32(false, x0, false, onesv,
                                                 (short)0, acc0, false, false);
    acc1 = __builtin_amdgcn_wmma_f32_16x16x4_f32(false, x1, false, onesv,
                                                 (short)0, acc1, false, false);
  }
  v8f acc = acc0 + acc1;
  float s = acc[0] + acc[1] + acc[2] + acc[3] + acc[4] + acc[5] + acc[6] + acc[7];
  s += __shfl_xor(s, 16, 32);
  __shared__ float wsum[8];
  if (lane == 0) wsum[wv] = s;
  __syncthreads();
  if (threadIdx.x == 0) {
    float tsum = 0.f;
    for (int w = 0; w < wpb; ++w) tsum += wsum[w];
    part[blockIdx.x] = tsum;
  }
}

// ---------------- final combine (single block, fixed-order => deterministic) ----------------
__global__ void k_final(const float* __restrict__ msep, const float* __restrict__ dfp,
                        int nparts, float* __restrict__ out, int n) {
  __shared__ float s1[THREADS];
  __shared__ float s2[THREADS];
  const int t = threadIdx.x;
  float a = 0.f, b = 0.f;
  for (int i = t; i < nparts; i += blockDim.x) { a += msep[i]; b += dfp[i]; }
  s1[t] = a; s2[t] = b;
  __syncthreads();
  for (int off = THREADS / 2; off > 0; off >>= 1) {
    if (t < off) { s1[t] += s1[t + off]; s2[t] += s2[t + off]; }
    __syncthreads();
  }
  if (t == 0) {
    float nn   = (float)n;
    float task = s1[0] / nn;
    float xi   = 1.0f - 3.0f * s2[0] / (nn * nn - 1.0f);
    out[0] = task - 1.0f * xi;   // LAMBDA = 1
    out[1] = xi;
  }
}

extern "C" void kernel_launch(void* const* d_in, const int* in_sizes, int n_in,
                              void* d_out, int out_size, void* d_ws, size_t ws_size,
                              hipStream_t stream) {
  (void)n_in; (void)out_size; (void)ws_size;
  const float* y_pred = (const float*)d_in[0];
  const float* y_true = (const float*)d_in[1];
  float* out = (float*)d_out;
  const int n    = in_sizes[0];
  const int nblk = n / TILE;

  char* w = (char*)d_ws;
  const size_t nb = (size_t)n * 4;
  float*    tn   = (float*)(w);
  uint32_t* keyA = (uint32_t*)(w + nb);
  uint32_t* valA = (uint32_t*)(w + 2 * nb);
  uint32_t* keyB = (uint32_t*)(w + 3 * nb);
  uint32_t* valB = (uint32_t*)(w + 4 * nb);
  uint32_t* hist = (uint32_t*)(w + 5 * nb);
  float* msep  = (float*)(w + 5 * nb + (size_t)256 * nblk * 4);
  float* diffp = msep + 256;

  const dim3 blk(THREADS);

  k_prep<<<512, blk, 0, stream>>>(y_true, tn, keyA, valA, n);

  // sort 1: (key(y_true+noise), index) -> perm in valA
  uint32_t *ki = keyA, *vi = valA, *ko = keyB, *vo = valB;
  for (int pass = 0; pass < 4; ++pass) {
    const int shift = pass * 8;
    k_hist<<<nblk, blk, 0, stream>>>(ki, hist, shift, nblk);
    k_scan<<<1, blk, 0, stream>>>(hist, nblk);
    k_scatter<<<nblk, blk, 0, stream>>>(ki, vi, ko, vo, hist, shift, nblk);
    uint32_t* t;
    t = ki; ki = ko; ko = t;
    t = vi; vi = vo; vo = t;
  }
  // 4 swaps -> sorted data back in keyA/valA

  k_mse<<<256, blk, 0, stream>>>(y_pred, tn, msep, n);

  k_gather<<<512, blk, 0, stream>>>(y_pred, keyA, valA, n);

  // sort 2: (key(y_pred_ord+noise), position) -> ranks
  for (int pass = 0; pass < 4; ++pass) {
    const int shift = pass * 8;
    k_hist<<<nblk, blk, 0, stream>>>(ki, hist, shift, nblk);
    k_scan<<<1, blk, 0, stream>>>(hist, nblk);
    k_scatter<<<nblk, blk, 0, stream>>>(ki, vi, ko, vo, hist, shift, nblk);
    uint32_t* t;
    t = ki; ki = ko; ko = t;
    t = vi; vi = vo; vo = t;
  }

  uint32_t* rk = keyB;   // free after sort 2
  k_rank<<<512, blk, 0, stream>>>(valA, rk, n);
  k_diff<<<256, blk, 0, stream>>>(rk, diffp, n);
  k_final<<<1, blk, 0, stream>>>(msep, diffp, 256, out, n);
}